// GraphNeuralNetwork_4818953306613
// MI455X (gfx1250) — compile-verified
//
#include <hip/hip_runtime.h>
#include <hip/hip_bf16.h>
#include <cmath>

typedef __attribute__((ext_vector_type(2))) float v2f;
typedef __attribute__((ext_vector_type(8))) float v8f;

#define DH 128          // hidden / input feature dim
#define NG 128          // number of graphs

// ---------- helpers: order-preserving float <-> uint for atomicMax ----------
__device__ __forceinline__ unsigned f2o(float f) {
    unsigned u = __float_as_uint(f);
    return (u >> 31) ? ~u : (u ^ 0x80000000u);
}
__device__ __forceinline__ float o2f(unsigned o) {
    return __uint_as_float((o >> 31) ? (o ^ 0x80000000u) : ~o);
}

// ---------- degree / normalization ----------
__global__ void k_deg_init(float* deg, int n) {
    int i = blockIdx.x * blockDim.x + threadIdx.x;
    if (i < n) deg[i] = 1.0f;                       // self-loop contribution
}
__global__ void k_deg_acc(const long long* __restrict__ ei, float* deg, int nedges) {
    int e = blockIdx.x * blockDim.x + threadIdx.x;
    if (e < nedges) {
        long long c = ei[(long long)nedges + e];    // col = edge_index[1][e]
        unsafeAtomicAdd(&deg[c], 1.0f);
    }
}
__global__ void k_dinv(float* deg, int n) {
    int i = blockIdx.x * blockDim.x + threadIdx.x;
    if (i < n) deg[i] = rsqrtf(deg[i]);             // deg >= 1 always (self loops)
}

// ---------- GEMM: t = act(hin) @ W ; hout = b + dinv^2 * t  (self-loop seed) ----------
// block = 256 threads (8 waves). Each wave owns a 16-row strip, all 128 columns
// as 8 16x16 WMMA tiles. W staged in LDS pre-swizzled in K-pairs so each
// B fragment is one contiguous ds_load_b64. Per k-quad: clause of 4
// ds_load_2addr_b64 (all 8 B fragments) -> one dscnt wait -> 8 WMMAs.
// K advanced 4 at a time with V_WMMA_F32_16X16X4_F32 (full fp32 precision).
template <bool TANH>
__global__ void k_gemm_seed(const float* __restrict__ hin, const float* __restrict__ W,
                            const float* __restrict__ bias, const float* __restrict__ dinv,
                            float* __restrict__ t, float* __restrict__ hout, int nnodes) {
    // Wl2[(k>>1)*128 + n] = { W[k][n], W[k+1][n] }   (k even)
    __shared__ v2f Wl2[(DH / 2) * DH];               // 64 KB
    const int tid   = threadIdx.x;
    const int wave  = tid >> 5;
    const int lane  = tid & 31;

    for (int i = tid; i < DH * DH; i += 256) {
        const int k = i >> 7;        // W row
        const int n = i & 127;       // W col
        Wl2[((k >> 1) << 7) + n][k & 1] = W[i];
    }
    __syncthreads();

    const int row0  = (blockIdx.x * 8 + wave) * 16;
    const int m     = lane & 15;        // A row / B col within tile
    const int khalf = lane >> 4;        // 0: K pair {0,1}, 1: K pair {2,3}
    const int r     = row0 + m;
    const int rc    = (r < nnodes) ? r : (nnodes - 1);   // clamp reads, mask writes

    v8f acc[8];
#pragma unroll
    for (int j = 0; j < 8; ++j) acc[j] = (v8f)(0.0f);

    for (int k0 = 0; k0 < DH; k0 += 4) {
        const int ka = k0 + khalf * 2;
        float ax = hin[rc * DH + ka];
        float ay = hin[rc * DH + ka + 1];
        if (TANH) { ax = tanhf(ax); ay = tanhf(ay); }
        v2f a; a.x = ax; a.y = ay;
        const int kp = (k0 >> 1) + khalf;            // == ka >> 1

        // phase 1: hoist all 8 B fragments (clause of 4 ds_load_2addr_b64)
        v2f bfrag[8];
#pragma unroll
        for (int j = 0; j < 8; ++j)
            bfrag[j] = Wl2[(kp << 7) + (j << 4) + m];
        // phase 2: 8 WMMAs, single dscnt wait up front
#pragma unroll
        for (int j = 0; j < 8; ++j)
            acc[j] = __builtin_amdgcn_wmma_f32_16x16x4_f32(
                false, a, false, bfrag[j], (short)0, acc[j], false, false);
    }

    const int n = lane & 15;
#pragma unroll
    for (int v = 0; v < 8; ++v) {
        const int rr = row0 + v + 8 * khalf;
        if (rr < nnodes) {
            const float d  = dinv[rr];
            const float dd = d * d;
#pragma unroll
            for (int j = 0; j < 8; ++j) {
                const int   c   = (j << 4) + n;
                const float val = acc[j][v];
                t[rr * DH + c]    = val;
                hout[rr * DH + c] = bias[c] + dd * val;   // bias + self-loop term
            }
        }
    }
}

// ---------- edge scatter: hout[col] += dinv[row]*dinv[col] * t[row] ----------
// one wave per edge; each lane moves 4 contiguous floats (32*4 = 128).
// t/hout are L2-resident (51 MB << 192 MB L2) -> atomics resolve in L2.
__global__ void k_scatter(const long long* __restrict__ ei, const float* __restrict__ t,
                          const float* __restrict__ dinv, float* __restrict__ hout,
                          int nedges) {
    const int e = blockIdx.x * 8 + (threadIdx.x >> 5);
    if (e >= nedges) return;
    const int lane = threadIdx.x & 31;
    const long long row = ei[e];
    const long long col = ei[(long long)nedges + e];
    const float norm = dinv[row] * dinv[col];
    const int c = lane * 4;
    const float4 v = *(const float4*)(t + row * DH + c);
    float* dst = hout + col * DH + c;
    unsafeAtomicAdd(dst + 0, v.x * norm);
    unsafeAtomicAdd(dst + 1, v.y * norm);
    unsafeAtomicAdd(dst + 2, v.z * norm);
    unsafeAtomicAdd(dst + 3, v.w * norm);
}

// ---------- pooling ----------
__global__ void k_pool_init(unsigned* pmax, float* psum, float* cnt) {
    int i = blockIdx.x * blockDim.x + threadIdx.x;
    if (i < NG * DH) { pmax[i] = 0u; psum[i] = 0.0f; }   // 0 == ordered(-inf-ish)
    if (i < NG) cnt[i] = 0.0f;
}
__global__ void k_pool(const float* __restrict__ h, const long long* __restrict__ batch,
                       unsigned* pmax, float* psum, float* cnt, int nnodes) {
    const int node = blockIdx.x;
    const int c    = threadIdx.x;        // 0..127
    if (node >= nnodes) return;
    const float v = tanhf(h[(long long)node * DH + c]);   // fused final activation
    const int g = (int)batch[node];
    atomicMax(&pmax[g * DH + c], f2o(v));
    unsafeAtomicAdd(&psum[g * DH + c], v);
    if (c == 0) unsafeAtomicAdd(&cnt[g], 1.0f);
}
__global__ void k_final(const unsigned* __restrict__ pmax, const float* __restrict__ psum,
                        const float* __restrict__ cnt, const float* __restrict__ Wl,
                        const float* __restrict__ bl, float* __restrict__ out) {
    const int g = threadIdx.x;           // 0..127, single block
    const float cg = cnt[g];
    float acc = bl[0];
    for (int c = 0; c < DH; ++c) {
        const float mx = (cg > 0.0f) ? o2f(pmax[g * DH + c]) : 0.0f;
        const float mn = psum[g * DH + c] / fmaxf(cg, 1.0f);
        acc += mx * Wl[c] + mn * Wl[DH + c];
    }
    out[g] = acc;
}

// ---------- host ----------
extern "C" void kernel_launch(void* const* d_in, const int* in_sizes, int n_in,
                              void* d_out, int out_size, void* d_ws, size_t ws_size,
                              hipStream_t stream) {
    const float*     x  = (const float*)d_in[0];
    const long long* ei = (const long long*)d_in[1];   // int64 (2, E)
    const long long* bi = (const long long*)d_in[2];   // int64 (N,)
    const float* W1 = (const float*)d_in[3];  const float* b1 = (const float*)d_in[4];
    const float* W2 = (const float*)d_in[5];  const float* b2 = (const float*)d_in[6];
    const float* W3 = (const float*)d_in[7];  const float* b3 = (const float*)d_in[8];
    const float* W4 = (const float*)d_in[9];  const float* b4 = (const float*)d_in[10];
    const float* Wl = (const float*)d_in[11]; const float* bl = (const float*)d_in[12];
    float* out = (float*)d_out;

    const int N = in_sizes[0] / DH;      // 100000
    const int E = in_sizes[1] / 2;       // 1600000

    // workspace layout (floats), 16B-aligned slices
    const size_t ndpad = ((size_t)N + 127) & ~(size_t)127;
    float*    dinv = (float*)d_ws;
    float*    t    = dinv + ndpad;
    float*    hA   = t  + (size_t)N * DH;
    float*    hB   = hA + (size_t)N * DH;
    unsigned* pmax = (unsigned*)(hB + (size_t)N * DH);
    float*    psum = (float*)(pmax + NG * DH);
    float*    cnt  = psum + NG * DH;

    const int B256 = 256;
    const dim3 gN((N + B256 - 1) / B256), gE((E + B256 - 1) / B256);
    const dim3 gGemm((N + 127) / 128);             // 8 waves * 16 rows per block
    const dim3 gScat((E + 7) / 8);                 // 8 edges (waves) per block

    // normalization
    k_deg_init<<<gN, B256, 0, stream>>>(dinv, N);
    k_deg_acc<<<gE, B256, 0, stream>>>(ei, dinv, E);
    k_dinv<<<gN, B256, 0, stream>>>(dinv, N);

    // layer 1: x -> hA
    k_gemm_seed<false><<<gGemm, B256, 0, stream>>>(x,  W1, b1, dinv, t, hA, N);
    k_scatter<<<gScat, B256, 0, stream>>>(ei, t, dinv, hA, E);
    // layer 2: tanh(hA) -> hB
    k_gemm_seed<true ><<<gGemm, B256, 0, stream>>>(hA, W2, b2, dinv, t, hB, N);
    k_scatter<<<gScat, B256, 0, stream>>>(ei, t, dinv, hB, E);
    // layer 3: tanh(hB) -> hA
    k_gemm_seed<true ><<<gGemm, B256, 0, stream>>>(hB, W3, b3, dinv, t, hA, N);
    k_scatter<<<gScat, B256, 0, stream>>>(ei, t, dinv, hA, E);
    // layer 4: tanh(hA) -> hB
    k_gemm_seed<true ><<<gGemm, B256, 0, stream>>>(hA, W4, b4, dinv, t, hB, N);
    k_scatter<<<gScat, B256, 0, stream>>>(ei, t, dinv, hB, E);

    // pooling + head (tanh fused into pool load)
    k_pool_init<<<(NG * DH + B256 - 1) / B256, B256, 0, stream>>>(pmax, psum, cnt);
    k_pool<<<N, DH, 0, stream>>>(hB, bi, pmax, psum, cnt, N);
    k_final<<<1, NG, 0, stream>>>(pmax, psum, cnt, Wl, bl, out);
}